// QuantumRegressor_65481071401095
// MI455X (gfx1250) — compile-verified
//
#include <hip/hip_runtime.h>
#include <hip/hip_bf16.h>

// QuantumRegressor on MI455X (gfx1250).
// Circuit of single-qubit gates tensor-factorizes: total U = V0 (x) V1 (x) ... (x) V15,
// V_w = U[layer1,w] @ U[layer0,w].  Grouped 4 wires at a time -> 16x16 complex matrices,
// applied with V_WMMA_F32_16X16X4_F32 (fp32 precision) on LDS-resident state quarters.

typedef float v2f __attribute__((ext_vector_type(2)));
typedef float v8f __attribute__((ext_vector_type(8)));

struct c32 { float r, i; };
__device__ __forceinline__ c32 cmul(c32 a, c32 b) {
  c32 o; o.r = a.r*b.r - a.i*b.i; o.i = a.r*b.i + a.i*b.r; return o;
}
__device__ __forceinline__ void mm2(c32 C[2][2], const c32 A[2][2], const c32 B[2][2]) {
  #pragma unroll
  for (int i = 0; i < 2; ++i)
    #pragma unroll
    for (int j = 0; j < 2; ++j) {
      c32 s = cmul(A[i][0], B[0][j]);
      c32 t = cmul(A[i][1], B[1][j]);
      C[i][j].r = s.r + t.r; C[i][j].i = s.i + t.i;
    }
}

// ws float layout:
//   g*768 + [0,256)  : Re(M_g)       (g=0: wires2-5, 1: wires6-9, 2: wires10-13, 3: I4(x)V14(x)V15)
//   g*768 + [256,512): Im(M_g)
//   g*768 + [512,768): -Im(M_g)      (f32 WMMA has no A-negate; NEG only supports CNeg)
//   [3072,3088): Re(V0 (x) V1)   [3088,3104): Im(V0 (x) V1)
__global__ void qr_setup(const float* __restrict__ params, float* __restrict__ ws) {
  __shared__ c32 Vs[16][2][2];
  int t = threadIdx.x;
  if (t < 16) {
    c32 U[2][2][2];
    for (int l = 0; l < 2; ++l) {
      float a = 0.5f * params[(l*16 + t)*3 + 0];
      float b = 0.5f * params[(l*16 + t)*3 + 1];
      float c = 0.5f * params[(l*16 + t)*3 + 2];
      float ca = cosf(a), sa = sinf(a);
      float cb = cosf(b), sb = sinf(b);
      float cc = cosf(c), sc = sinf(c);
      c32 RX[2][2] = {{{ca,0.f},{0.f,-sa}},{{0.f,-sa},{ca,0.f}}};
      c32 RY[2][2] = {{{cb,0.f},{-sb,0.f}},{{sb,0.f},{cb,0.f}}};
      c32 RZ[2][2] = {{{cc,-sc},{0.f,0.f}},{{0.f,0.f},{cc,sc}}};
      c32 T[2][2];
      mm2(T, RY, RX);
      mm2(U[l], RZ, T);
    }
    c32 V[2][2];
    mm2(V, U[1], U[0]);   // layer0 applied first => V = U1 * U0
    for (int i = 0; i < 2; ++i)
      for (int j = 0; j < 2; ++j) Vs[t][i][j] = V[i][j];
  }
  __syncthreads();
  int i = t >> 4, j = t & 15;
  for (int g = 0; g < 3; ++g) {          // wires 2-5, 6-9, 10-13 (MSB-first kron)
    int w0 = 2 + 4*g;
    c32 e = {1.f, 0.f};
    for (int s = 0; s < 4; ++s)
      e = cmul(e, Vs[w0 + s][(i >> (3 - s)) & 1][(j >> (3 - s)) & 1]);
    ws[g*768 + t]       = e.r;
    ws[g*768 + 256 + t] = e.i;
    ws[g*768 + 512 + t] = -e.i;
  }
  // M3 = I4 (x) (V14 (x) V15): 16-axis covers bits (b12 b13 | b14 b15)
  c32 e3 = {0.f, 0.f};
  if ((i >> 2) == (j >> 2))
    e3 = cmul(Vs[14][(i >> 1) & 1][(j >> 1) & 1], Vs[15][i & 1][j & 1]);
  ws[2304 + t] = e3.r; ws[2304 + 256 + t] = e3.i; ws[2304 + 512 + t] = -e3.i;
  if (t < 16) {                           // W4 = V0 (x) V1 (4x4, used at load time)
    int qi = t >> 2, qj = t & 3;
    c32 w = cmul(Vs[0][qi >> 1][qj >> 1], Vs[1][qi & 1][qj & 1]);
    ws[3072 + t] = w.r; ws[3088 + t] = w.i;
  }
}

__global__ void qr_init(float* __restrict__ out, const float* __restrict__ head_b, int B) {
  int idx = blockIdx.x * blockDim.x + threadIdx.x;
  if (idx < B) out[idx] = head_b[0];
}

__global__ __launch_bounds__(256)
void qr_main(const float* __restrict__ x, const float* __restrict__ ws,
             const float* __restrict__ head_w, float* __restrict__ out) {
  extern __shared__ float lds[];
  float* sr  = lds;            // 16384 floats: Re(state quarter)
  float* si  = lds + 16384;    // 16384 floats: Im(state quarter)
  float* red = lds + 32768;    // [0]=||x||^2, [1..16]=signed prob sums per wire
  const int tid = threadIdx.x;
  const int n = blockIdx.x >> 2;     // batch element
  const int q = blockIdx.x & 3;      // value of (wire0,wire1) bits owned by this WG
  if (tid < 17) red[tid] = 0.f;

  float w4r[4], w4i[4];
  #pragma unroll
  for (int qp = 0; qp < 4; ++qp) {
    w4r[qp] = ws[3072 + q*4 + qp];
    w4i[qp] = ws[3088 + q*4 + qp];
  }
  __syncthreads();

  // ---- load pass: read full element (real), apply V0(x)V1, accumulate ||x||^2 ----
  const float4* x4 = reinterpret_cast<const float4*>(x + (size_t)n * 65536);
  float nrm = 0.f;
  for (int it = 0; it < 16; ++it) {
    int f = tid + (it << 8);          // float4 index within a 16384-float quarter
    float4 ar = {0.f,0.f,0.f,0.f}, ai = {0.f,0.f,0.f,0.f};
    #pragma unroll
    for (int qp = 0; qp < 4; ++qp) {
      float4 v = x4[qp*4096 + f];
      nrm += v.x*v.x + v.y*v.y + v.z*v.z + v.w*v.w;
      float wr = w4r[qp], wi = w4i[qp];
      ar.x += wr*v.x; ar.y += wr*v.y; ar.z += wr*v.z; ar.w += wr*v.w;
      ai.x += wi*v.x; ai.y += wi*v.y; ai.z += wi*v.z; ai.w += wi*v.w;
    }
    int jj = f << 2;
    sr[jj] = ar.x; sr[jj+1] = ar.y; sr[jj+2] = ar.z; sr[jj+3] = ar.w;
    si[jj] = ai.x; si[jj+1] = ai.y; si[jj+2] = ai.z; si[jj+3] = ai.w;
  }
  #pragma unroll
  for (int off = 16; off > 0; off >>= 1) nrm += __shfl_xor(nrm, off, 32);
  if ((tid & 31) == 0) atomicAdd(&red[0], nrm);
  __syncthreads();

  // ---- 4 WMMA passes: 16x16 complex matrix along one axis of [L,16,R] ----
  const int wave = tid >> 5, ln = tid & 31;
  const int hi = ln >> 4, m = ln & 15;
  const int lrs[4] = {10, 6, 2, 0};   // log2(R) per pass
  #pragma unroll
  for (int pass = 0; pass < 4; ++pass) {
    const int lr = lrs[pass];
    const float* Mr = ws + pass*768;
    const float* Mi = Mr + 256;
    const float* Mn = Mr + 512;       // -Im(M)
    v2f Ar[4], Ai[4], An[4];          // A frag: lane m, VGPR v holds M[m, 4k+v+2*hi]
    #pragma unroll
    for (int k = 0; k < 4; ++k) {
      int cb = m*16 + 4*k + 2*hi;
      Ar[k][0] = Mr[cb]; Ar[k][1] = Mr[cb+1];
      Ai[k][0] = Mi[cb]; Ai[k][1] = Mi[cb+1];
      An[k][0] = Mn[cb]; An[k][1] = Mn[cb+1];
    }
    for (int t8 = 0; t8 < 8; ++t8) {       // 64 units / 8 waves
      int u = wave*8 + t8;
      int c = u*16 + m;                    // column id = l*R + r
      int base = ((c >> lr) << (lr + 4)) | (c & ((1 << lr) - 1));
      v2f Br[4], Bi[4];                    // B frag: VGPR v holds row 4k+v+2*hi, col m
      #pragma unroll
      for (int k = 0; k < 4; ++k) {
        int j0 = 4*k + 2*hi;
        Br[k][0] = sr[base + ( j0      << lr)];
        Br[k][1] = sr[base + ((j0 + 1) << lr)];
        Bi[k][0] = si[base + ( j0      << lr)];
        Bi[k][1] = si[base + ((j0 + 1) << lr)];
      }
      v8f Dr = {0.f,0.f,0.f,0.f,0.f,0.f,0.f,0.f};
      v8f Di = {0.f,0.f,0.f,0.f,0.f,0.f,0.f,0.f};
      #pragma unroll
      for (int k = 0; k < 4; ++k) {        // Dr = Mr*Sr - Mi*Si ; Di = Mr*Si + Mi*Sr
        Dr = __builtin_amdgcn_wmma_f32_16x16x4_f32(false, Ar[k], false, Br[k], (short)0, Dr, false, false);
        Dr = __builtin_amdgcn_wmma_f32_16x16x4_f32(false, An[k], false, Bi[k], (short)0, Dr, false, false);
        Di = __builtin_amdgcn_wmma_f32_16x16x4_f32(false, Ar[k], false, Bi[k], (short)0, Di, false, false);
        Di = __builtin_amdgcn_wmma_f32_16x16x4_f32(false, Ai[k], false, Br[k], (short)0, Di, false, false);
      }
      #pragma unroll
      for (int v = 0; v < 8; ++v) {        // D frag: VGPR v = row v+8*hi, col m
        int addr = base + ((v + 8*hi) << lr);
        sr[addr] = Dr[v];
        si[addr] = Di[v];
      }
    }
    __syncthreads();
  }

  // ---- features: <Z_w> signed probability sums, then head dot product ----
  float feat[16];
  #pragma unroll
  for (int w = 0; w < 16; ++w) feat[w] = 0.f;
  for (int it = 0; it < 64; ++it) {
    int jj = tid + (it << 8);
    float a = sr[jj], b = si[jj];
    float p = a*a + b*b;
    feat[0] += (q & 2) ? -p : p;          // wire0 = MSB of full index
    feat[1] += (q & 1) ? -p : p;
    #pragma unroll
    for (int w = 2; w < 16; ++w)
      feat[w] += ((jj >> (15 - w)) & 1) ? -p : p;
  }
  #pragma unroll
  for (int w = 0; w < 16; ++w) {
    float v = feat[w];
    #pragma unroll
    for (int off = 16; off > 0; off >>= 1) v += __shfl_xor(v, off, 32);
    if (ln == 0) atomicAdd(&red[1 + w], v);
  }
  __syncthreads();
  if (tid == 0) {
    float inv = 1.f / red[0];             // deferred normalization: p scales by 1/||x||^2
    float acc = 0.f;
    #pragma unroll
    for (int w = 0; w < 16; ++w) acc += head_w[w] * red[1 + w];
    atomicAdd(&out[n], acc * inv);
  }
}

extern "C" void kernel_launch(void* const* d_in, const int* in_sizes, int n_in,
                              void* d_out, int out_size, void* d_ws, size_t ws_size,
                              hipStream_t stream) {
  const float* x      = (const float*)d_in[0];
  const float* params = (const float*)d_in[1];
  const float* head_w = (const float*)d_in[2];
  const float* head_b = (const float*)d_in[3];
  float* out = (float*)d_out;
  float* ws  = (float*)d_ws;
  int B = in_sizes[0] / 65536;
  size_t shmem = (size_t)(32768 + 32) * sizeof(float);  // 128KB state + reduction slots
  (void)hipFuncSetAttribute(reinterpret_cast<const void*>(qr_main),
                            hipFuncAttributeMaxDynamicSharedMemorySize, (int)shmem);
  qr_init<<<(B + 255) / 256, 256, 0, stream>>>(out, head_b, B);
  qr_setup<<<1, 256, 0, stream>>>(params, ws);
  qr_main<<<B * 4, 256, shmem, stream>>>(x, ws, head_w, out);
}